// NeuMFModel_80693845557568
// MI455X (gfx1250) — compile-verified
//
#include <hip/hip_runtime.h>
#include <hip/hip_bf16.h>

typedef float v2f __attribute__((ext_vector_type(2)));
typedef float v8f __attribute__((ext_vector_type(8)));

#define LDS_S 17  // stride padding: 17*r mod 64 distinct for r=0..15 -> conflict-free

__global__ __launch_bounds__(256) void neumf_fused(
    const int*   __restrict__ user_ids,
    const int*   __restrict__ movie_ids,
    const float* __restrict__ gmf_user,
    const float* __restrict__ gmf_movie,
    const float* __restrict__ mlp_user,
    const float* __restrict__ mlp_movie,
    const float* __restrict__ W1,
    const float* __restrict__ b1,
    const float* __restrict__ W2,
    const float* __restrict__ b2,
    const float* __restrict__ Wf,
    const float* __restrict__ bf,
    float*       __restrict__ out)
{
    __shared__ float h1s[8][16 * LDS_S];
    __shared__ float h2s[8][16 * LDS_S];

    const int tid  = threadIdx.x;
    const int wave = tid >> 5;   // 8 waves per block
    const int lane = tid & 31;   // wave32
    const int r16  = lane & 15;  // row within 16-row tile
    const int hi   = lane >> 4;  // half-wave id

    const int row = (blockIdx.x * 8 + wave) * 16 + r16;
    const int uid = user_ids[row];
    const int mid = movie_ids[row];

    // ---------------- GMF branch fused with final projection ----------------
    // combined@Wf over GMF part: per-lane partial dot across 32 of 64 columns.
    const float4* gu = (const float4*)(gmf_user  + (size_t)uid * 64 + hi * 32);
    const float4* gm = (const float4*)(gmf_movie + (size_t)mid * 64 + hi * 32);
    const float4* wf = (const float4*)(Wf + hi * 32);
    float part = 0.f;
#pragma unroll
    for (int i = 0; i < 8; ++i) {
        float4 a = gu[i]; float4 b = gm[i]; float4 w = wf[i];
        part = fmaf(a.x * b.x, w.x, part);
        part = fmaf(a.y * b.y, w.y, part);
        part = fmaf(a.z * b.z, w.z, part);
        part = fmaf(a.w * b.w, w.w, part);
    }
    const float gmf_dot = part + __shfl_xor(part, 16, 32);

    // ---------------- MLP layer 1: [16x16] @ W1(16x8 pad 16) via 4x WMMA K=4 ----
    // A layout (16x4 f32): lane l holds M=l%16, VGPR v -> K = 2*(l/16)+v.
    const float2* mu = (const float2*)(mlp_user  + (size_t)uid * 8);
    const float2* mm = (const float2*)(mlp_movie + (size_t)mid * 8);
    float2 u0 = mu[hi],     u1 = mu[2 + hi];
    float2 m0 = mm[hi],     m1 = mm[2 + hi];
    v2f a0 = {u0.x, u0.y};             // K = 0..3   (mlp_in cols 0..7 = user emb)
    v2f a1 = {u1.x, u1.y};             // K = 4..7
    v2f a2 = {m0.x, m0.y};             // K = 8..11  (cols 8..15 = movie emb)
    v2f a3 = {m1.x, m1.y};             // K = 12..15

    // B layout (4x16 f32): lane l holds N=l%16, VGPR v -> K = 2*(l/16)+v.
    const int   n1 = (r16 < 8) ? r16 : 0;
    const float k1 = (r16 < 8) ? 1.f : 0.f;   // zero-pad N=8..15
    const int   kb = 2 * hi;
    v2f b0c = { W1[(kb + 0) * 8 + n1] * k1,  W1[(kb + 1) * 8 + n1] * k1 };
    v2f b1c = { W1[(4 + kb) * 8 + n1] * k1,  W1[(5 + kb) * 8 + n1] * k1 };
    v2f b2c = { W1[(8 + kb) * 8 + n1] * k1,  W1[(9 + kb) * 8 + n1] * k1 };
    v2f b3c = { W1[(12 + kb) * 8 + n1] * k1, W1[(13 + kb) * 8 + n1] * k1 };

    const float cb1 = b1[n1] * k1;     // bias broadcast down M
    v8f acc = {cb1, cb1, cb1, cb1, cb1, cb1, cb1, cb1};
    acc = __builtin_amdgcn_wmma_f32_16x16x4_f32(false, a0, false, b0c, (short)0, acc, false, false);
    acc = __builtin_amdgcn_wmma_f32_16x16x4_f32(false, a1, false, b1c, (short)0, acc, false, false);
    acc = __builtin_amdgcn_wmma_f32_16x16x4_f32(false, a2, false, b2c, (short)0, acc, false, false);
    acc = __builtin_amdgcn_wmma_f32_16x16x4_f32(false, a3, false, b3c, (short)0, acc, false, false);
#pragma unroll
    for (int v = 0; v < 8; ++v) acc[v] = fmaxf(acc[v], 0.f);   // ReLU

    // D layout -> LDS [batch_row][feature] for the layer-2 A-layout reload.
    float* h1 = h1s[wave];
#pragma unroll
    for (int v = 0; v < 8; ++v) h1[(v + 8 * hi) * LDS_S + r16] = acc[v];
    asm volatile("s_wait_dscnt 0" ::: "memory");   // wave-local DS ordering fence

    // ---------------- MLP layer 2: [16x8] @ W2(8x4 pad 16) via 2x WMMA K=4 -----
    v2f p0 = { h1[r16 * LDS_S + kb + 0], h1[r16 * LDS_S + kb + 1] };
    v2f p1 = { h1[r16 * LDS_S + 4 + kb], h1[r16 * LDS_S + 5 + kb] };

    const int   n2 = (r16 < 4) ? r16 : 0;
    const float k2 = (r16 < 4) ? 1.f : 0.f;   // zero-pad N=4..15
    v2f q0 = { W2[(kb + 0) * 4 + n2] * k2, W2[(kb + 1) * 4 + n2] * k2 };
    v2f q1 = { W2[(4 + kb) * 4 + n2] * k2, W2[(5 + kb) * 4 + n2] * k2 };

    const float cb2 = b2[n2] * k2;
    v8f acc2 = {cb2, cb2, cb2, cb2, cb2, cb2, cb2, cb2};
    acc2 = __builtin_amdgcn_wmma_f32_16x16x4_f32(false, p0, false, q0, (short)0, acc2, false, false);
    acc2 = __builtin_amdgcn_wmma_f32_16x16x4_f32(false, p1, false, q1, (short)0, acc2, false, false);
#pragma unroll
    for (int v = 0; v < 8; ++v) acc2[v] = fmaxf(acc2[v], 0.f); // ReLU

    float* h2 = h2s[wave];
#pragma unroll
    for (int v = 0; v < 8; ++v) h2[(v + 8 * hi) * LDS_S + r16] = acc2[v];
    asm volatile("s_wait_dscnt 0" ::: "memory");

    // ---------------- Final projection tail + sigmoid*4+1 ----------------
    const float mlp_dot = h2[r16 * LDS_S + 0] * Wf[64]
                        + h2[r16 * LDS_S + 1] * Wf[65]
                        + h2[r16 * LDS_S + 2] * Wf[66]
                        + h2[r16 * LDS_S + 3] * Wf[67];

    const float logit = gmf_dot + mlp_dot + bf[0];
    const float res = 4.f * (1.f / (1.f + __expf(-logit))) + 1.f;
    if (hi == 0) out[row] = res;
}

extern "C" void kernel_launch(void* const* d_in, const int* in_sizes, int n_in,
                              void* d_out, int out_size, void* d_ws, size_t ws_size,
                              hipStream_t stream) {
    const int B = in_sizes[0];               // 16384
    dim3 grid(B / 128), block(256);          // 8 waves/block * 16 rows/wave
    neumf_fused<<<grid, block, 0, stream>>>(
        (const int*)d_in[0],  (const int*)d_in[1],
        (const float*)d_in[2], (const float*)d_in[3],
        (const float*)d_in[4], (const float*)d_in[5],
        (const float*)d_in[6], (const float*)d_in[7],
        (const float*)d_in[8], (const float*)d_in[9],
        (const float*)d_in[10], (const float*)d_in[11],
        (float*)d_out);
}